// GAT_DGL_58110907515580
// MI455X (gfx1250) — compile-verified
//
#include <hip/hip_runtime.h>
#include <math.h>

// ---------------- problem constants (match reference) ----------------
#define N_NODES   50000
#define N_EDGES   800000
#define IN_DIM    256
#define HID       32
#define HEADS     8
#define N_CLASSES 40
#define NEG_SLOPE 0.2f

typedef float v2f __attribute__((ext_vector_type(2)));
typedef float v8f __attribute__((ext_vector_type(8)));

__device__ __forceinline__ float lrelu(float v) { return v > 0.f ? v : NEG_SLOPE * v; }
__device__ __forceinline__ float fix_inf(float m) { return (m == -__builtin_inff()) ? 0.f : m; }

// float atomic max via signed/unsigned integer ordering trick.
// Valid given emax initialized to -inf (0xFF800000: very negative as int, huge as uint).
__device__ __forceinline__ void atomicMaxF(float* addr, float v) {
  if (v >= 0.f) atomicMax((int*)addr, __float_as_int(v));
  else          atomicMin((unsigned int*)addr, __float_as_uint(v));
}

// ---------------- generic fill ----------------
__global__ void fill_f32(float* __restrict__ p, float v, int n) {
  int i = blockIdx.x * blockDim.x + threadIdx.x;
  if (i < n) p[i] = v;
}

// ---------------- fp32 WMMA GEMM: C[M,N] = op(A)[M,K] * B[K,N], row-major ----------------
// One wave (32 lanes) computes one 16x16 tile of C via V_WMMA_F32_16X16X4_F32.
// M must be a multiple of 16 (50000 = 16*3125); K a multiple of 4.
// RELU: apply ReLU to A elements on load (fuses layer-1 activation into layer-2 GEMM).
// GUARD: branchless handling of N not a multiple of 16 (clamp column + 0/1 mask on B),
//        keeping EXEC all-ones through the whole K-loop as WMMA requires.
template <bool RELU, bool GUARD>
__global__ void wmma_gemm_f32(const float* __restrict__ A, const float* __restrict__ B,
                              float* __restrict__ C, int N, int K) {
  const int lane = threadIdx.x;        // 0..31, wave32
  const int half = lane >> 4;          // 0 or 1
  const int l15  = lane & 15;
  const int rowTile = blockIdx.y << 4;
  const int colTile = blockIdx.x << 4;

  int   col  = colTile + l15;
  float mask = 1.f;
  if (GUARD) {
    if (col >= N) { mask = 0.f; col = N - 1; }  // loads stay in-bounds; values zeroed
  }

  v8f acc = {};
  // A layout (16x4 f32): lanes 0-15 rows 0-15 with K={k0,k0+1}; lanes 16-31 K={k0+2,k0+3}
  const float* ap = A + (size_t)(rowTile + l15) * K + 2 * half;
  // B layout (4x16 f32): lanes 0-15 cols with K={k0,k0+1}; lanes 16-31 K={k0+2,k0+3}
  const float*  bp    = B + (size_t)(2 * half) * N + col;
  const size_t  bstep = (size_t)4 * N;

  for (int k0 = 0; k0 < K; k0 += 4) {
    v2f a;
    a.x = ap[0];
    a.y = ap[1];
    if (RELU) { a.x = fmaxf(a.x, 0.f); a.y = fmaxf(a.y, 0.f); }

    v2f b;
    b.x = bp[0];
    b.y = bp[N];
    if (GUARD) { b.x *= mask; b.y *= mask; }

    // (neg_a, A, neg_b, B, c_mod, C, reuse_a, reuse_b)
    acc = __builtin_amdgcn_wmma_f32_16x16x4_f32(false, a, false, b, (short)0, acc,
                                                false, false);
    ap += 4;
    bp += bstep;
  }

  // C/D layout: VGPR i -> row = rowTile + i + 8*half, col = colTile + (lane&15)
  const int ocol = colTile + l15;
  if (!GUARD || ocol < N) {
    float* crow = C + (size_t)(rowTile + half * 8) * N + ocol;
#pragma unroll
    for (int i = 0; i < 8; i++) crow[(size_t)i * N] = acc[i];
  }
}

// ---------------- per-node attention logits: el/er = sum_d feat*al / feat*ar ----------------
__global__ void attn_coef(const float* __restrict__ feat, const float* __restrict__ al,
                          const float* __restrict__ ar, float* __restrict__ el,
                          float* __restrict__ er, int heads, int dout) {
  int i = blockIdx.x * blockDim.x + threadIdx.x;
  if (i >= N_NODES * heads) return;
  int n = i / heads, h = i - n * heads;
  const float* f   = feat + (size_t)n * heads * dout + (size_t)h * dout;
  const float* alh = al + (size_t)h * dout;
  const float* arh = ar + (size_t)h * dout;
  float sl = 0.f, sr = 0.f;
  for (int d = 0; d < dout; d++) { sl += f[d] * alh[d]; sr += f[d] * arh[d]; }
  el[i] = sl;
  er[i] = sr;
}

// ---------------- edge pass 1: segment max of leaky_relu(el[src]+er[dst]) ----------------
__global__ void edge_max(const int* __restrict__ src, const int* __restrict__ dst,
                         const float* __restrict__ el, const float* __restrict__ er,
                         float* __restrict__ emax, int heads) {
  int i = blockIdx.x * blockDim.x + threadIdx.x;
  if (i >= N_EDGES * heads) return;
  int e = i / heads, h = i - e * heads;
  int s = src[e], d = dst[e];
  float v = lrelu(el[s * heads + h] + er[d * heads + h]);
  atomicMaxF(&emax[d * heads + h], v);
}

// ---------------- edge pass 2: denom = segment sum of exp(e - emax[dst]) ----------------
__global__ void edge_sum(const int* __restrict__ src, const int* __restrict__ dst,
                         const float* __restrict__ el, const float* __restrict__ er,
                         const float* __restrict__ emax, float* __restrict__ denom,
                         int heads) {
  int i = blockIdx.x * blockDim.x + threadIdx.x;
  if (i >= N_EDGES * heads) return;
  int e = i / heads, h = i - e * heads;
  int s = src[e], d = dst[e];
  float v = lrelu(el[s * heads + h] + er[d * heads + h]);
  float m = fix_inf(emax[d * heads + h]);
  atomicAdd(&denom[d * heads + h], expf(v - m));
}

// ---------------- edge pass 3: out[dst] += alpha * feat[src] ----------------
__global__ void edge_aggregate(const int* __restrict__ src, const int* __restrict__ dst,
                               const float* __restrict__ el, const float* __restrict__ er,
                               const float* __restrict__ emax, const float* __restrict__ denom,
                               const float* __restrict__ feat, float* __restrict__ out,
                               int heads, int dout) {
  int i = blockIdx.x * blockDim.x + threadIdx.x;
  if (i >= N_EDGES * heads) return;
  int e = i / heads, h = i - e * heads;
  int s = src[e], d = dst[e];
  float v = lrelu(el[s * heads + h] + er[d * heads + h]);
  float m = fix_inf(emax[d * heads + h]);
  float alpha = expf(v - m) / denom[d * heads + h];
  const float* fs = feat + (size_t)s * heads * dout + (size_t)h * dout;
  float* op = out + (size_t)d * heads * dout + (size_t)h * dout;
  for (int k = 0; k < dout; k++) atomicAdd(&op[k], fs[k] * alpha);
}

// ---------------- row-wise log_softmax, in place ----------------
__global__ void log_softmax_rows(float* __restrict__ out) {
  int n = blockIdx.x * blockDim.x + threadIdx.x;
  if (n >= N_NODES) return;
  float* row = out + (size_t)n * N_CLASSES;
  float m = -__builtin_inff();
  for (int c = 0; c < N_CLASSES; c++) m = fmaxf(m, row[c]);
  float s = 0.f;
  for (int c = 0; c < N_CLASSES; c++) s += expf(row[c] - m);
  float ls = logf(s) + m;
  for (int c = 0; c < N_CLASSES; c++) row[c] = row[c] - ls;
}

// ---------------- host orchestration ----------------
static inline int cdiv(int a, int b) { return (a + b - 1) / b; }

extern "C" void kernel_launch(void* const* d_in, const int* in_sizes, int n_in,
                              void* d_out, int out_size, void* d_ws, size_t ws_size,
                              hipStream_t stream) {
  const float* x   = (const float*)d_in[0];
  const int*   src = (const int*)  d_in[1];
  const int*   dst = (const int*)  d_in[2];
  const float* W1  = (const float*)d_in[3];
  const float* al1 = (const float*)d_in[4];
  const float* ar1 = (const float*)d_in[5];
  const float* W2  = (const float*)d_in[6];
  const float* al2 = (const float*)d_in[7];
  const float* ar2 = (const float*)d_in[8];
  float* out = (float*)d_out;

  // workspace carve-up (~118 MB total; fits in the 192 MB L2 for the edge phase)
  char* ws = (char*)d_ws;
  size_t off = 0;
  auto alloc = [&](size_t nfloats) {
    float* p = (float*)(ws + off);
    off += nfloats * sizeof(float);
    return p;
  };
  float* feat1  = alloc((size_t)N_NODES * HEADS * HID);   // x @ W1     [N, 8, 32]
  float* agg1   = alloc((size_t)N_NODES * HEADS * HID);   // layer-1 aggregate
  float* el1    = alloc((size_t)N_NODES * HEADS);
  float* er1    = alloc((size_t)N_NODES * HEADS);
  float* emax1  = alloc((size_t)N_NODES * HEADS);
  float* denom1 = alloc((size_t)N_NODES * HEADS);
  float* feat2  = alloc((size_t)N_NODES * N_CLASSES);     // relu(agg1) @ W2
  float* el2    = alloc((size_t)N_NODES);
  float* er2    = alloc((size_t)N_NODES);
  float* emax2  = alloc((size_t)N_NODES);
  float* denom2 = alloc((size_t)N_NODES);

  const int BT = 256;
  const float NEG_INF = -__builtin_inff();

  // ===== Layer 1 =====
  // feat1 = x @ W1 : 3125 row tiles x 16 col tiles, one wave per tile; clean inner loop
  wmma_gemm_f32<false, false><<<dim3(IN_DIM / 16, N_NODES / 16), 32, 0, stream>>>(
      x, W1, feat1, HEADS * HID, IN_DIM);

  attn_coef<<<cdiv(N_NODES * HEADS, BT), BT, 0, stream>>>(feat1, al1, ar1, el1, er1,
                                                          HEADS, HID);

  fill_f32<<<cdiv(N_NODES * HEADS, BT), BT, 0, stream>>>(emax1, NEG_INF, N_NODES * HEADS);
  fill_f32<<<cdiv(N_NODES * HEADS, BT), BT, 0, stream>>>(denom1, 0.f, N_NODES * HEADS);

  edge_max<<<cdiv(N_EDGES * HEADS, BT), BT, 0, stream>>>(src, dst, el1, er1, emax1, HEADS);
  edge_sum<<<cdiv(N_EDGES * HEADS, BT), BT, 0, stream>>>(src, dst, el1, er1, emax1, denom1,
                                                         HEADS);

  fill_f32<<<cdiv(N_NODES * HEADS * HID, BT), BT, 0, stream>>>(agg1, 0.f,
                                                               N_NODES * HEADS * HID);
  edge_aggregate<<<cdiv(N_EDGES * HEADS, BT), BT, 0, stream>>>(
      src, dst, el1, er1, emax1, denom1, feat1, agg1, HEADS, HID);

  // ===== Layer 2 =====
  // feat2 = relu(agg1) @ W2 (ReLU fused into A-load); N=40 handled branchlessly
  wmma_gemm_f32<true, true><<<dim3(cdiv(N_CLASSES, 16), N_NODES / 16), 32, 0, stream>>>(
      agg1, W2, feat2, N_CLASSES, HEADS * HID);

  attn_coef<<<cdiv(N_NODES, BT), BT, 0, stream>>>(feat2, al2, ar2, el2, er2, 1, N_CLASSES);

  fill_f32<<<cdiv(N_NODES, BT), BT, 0, stream>>>(emax2, NEG_INF, N_NODES);
  fill_f32<<<cdiv(N_NODES, BT), BT, 0, stream>>>(denom2, 0.f, N_NODES);

  edge_max<<<cdiv(N_EDGES, BT), BT, 0, stream>>>(src, dst, el2, er2, emax2, 1);
  edge_sum<<<cdiv(N_EDGES, BT), BT, 0, stream>>>(src, dst, el2, er2, emax2, denom2, 1);

  fill_f32<<<cdiv(N_NODES * N_CLASSES, BT), BT, 0, stream>>>(out, 0.f, N_NODES * N_CLASSES);
  edge_aggregate<<<cdiv(N_EDGES, BT), BT, 0, stream>>>(src, dst, el2, er2, emax2, denom2,
                                                       feat2, out, 1, N_CLASSES);

  // mean over 1 head is identity; finish with in-place log_softmax
  log_softmax_rows<<<cdiv(N_NODES, BT), BT, 0, stream>>>(out);
}